// SparseMHA_outdegree_78623671321111
// MI455X (gfx1250) — compile-verified
//
#include <hip/hip_runtime.h>
#include <hip/hip_bf16.h>

typedef __attribute__((ext_vector_type(2))) float v2f;
typedef __attribute__((ext_vector_type(8))) float v8f;

#define NNODES 50000
#define DEG    16
#define HID    128
#define NHEADS 8
#define HDIM   16
#define NTILES (NNODES / 16)   // 3125, exact

// ---------------------------------------------------------------------------
// Kernel A: fused QKV projection  out = (h @ W^T + b) * scale
// One wave (32 threads) computes a 16-row x 128-col tile with
// V_WMMA_F32_16X16X4_F32. grid.y selects q/k/v.
// Output is stored head-major: outT[(head*N + node)*16 + d]  (col = d*8+head)
// so the edge kernel can gather with contiguous float4 loads.
// ---------------------------------------------------------------------------
__global__ __launch_bounds__(32)
void qkv_project_wmma(const float* __restrict__ h,
                      const float* __restrict__ Wq, const float* __restrict__ bq,
                      const float* __restrict__ Wk, const float* __restrict__ bk,
                      const float* __restrict__ Wv, const float* __restrict__ bv,
                      float* __restrict__ qkvT) {
  const int tile  = blockIdx.x;        // 0..NTILES-1
  const int which = blockIdx.y;        // 0:q 1:k 2:v

  const float* W;  const float* bias;  float scale;
  if (which == 0)      { W = Wq; bias = bq; scale = 0.25f; }  // head_dim^-0.5
  else if (which == 1) { W = Wk; bias = bk; scale = 1.0f;  }
  else                 { W = Wv; bias = bv; scale = 1.0f;  }
  float* out = qkvT + (size_t)which * NNODES * HID;

  const int lane  = threadIdx.x;       // 0..31
  const int mrow  = lane & 15;         // A-row / B-col / C-col index per lane
  const int khalf = (lane >> 4) << 1;  // lanes 0-15 -> K=0,1 ; lanes 16-31 -> K=2,3
  const int row0  = tile * 16;

  v8f acc[8];
  #pragma unroll
  for (int nt = 0; nt < 8; ++nt) acc[nt] = (v8f)(0.0f);

  const float* arow = h + (size_t)(row0 + mrow) * HID + khalf;
  for (int kk = 0; kk < HID; kk += 4) {
    // A fragment: 16x4 fp32, lane holds A[mrow, khalf:khalf+2]
    v2f a = *(const v2f*)(arow + kk);
    #pragma unroll
    for (int nt = 0; nt < 8; ++nt) {
      // B = W^T (4x16): lane holds B[khalf:khalf+2, col] = W[col, kk+khalf ...]
      const int col = nt * 16 + mrow;
      v2f b = *(const v2f*)(W + (size_t)col * HID + kk + khalf);
      acc[nt] = __builtin_amdgcn_wmma_f32_16x16x4_f32(
          /*neg_a=*/false, a, /*neg_b=*/false, b,
          /*c_mod=*/(short)0, acc[nt], /*reuse_a=*/false, /*reuse_b=*/false);
    }
  }

  // C layout: lanes 0-15 hold rows 0..7 (vgpr r -> M=r), lanes 16-31 rows 8..15
  const int rbase = row0 + ((lane >> 4) << 3);
  #pragma unroll
  for (int nt = 0; nt < 8; ++nt) {
    const int col  = nt * 16 + mrow;   // flat element = d*8 + head
    const int head = col & 7;
    const int d    = col >> 3;
    const float bc = bias[col];
    float* obase = out + (size_t)head * NNODES * HDIM + d;
    #pragma unroll
    for (int r = 0; r < 8; ++r) {
      obase[(size_t)(rbase + r) * HDIM] = (acc[nt][r] + bc) * scale;
    }
  }
}

// ---------------------------------------------------------------------------
// Kernel B: fused SDDMM + segment-softmax + SPMM.
// One 128-thread block per row. Half-wave (16 lanes) = one head's 16 edges:
//   thread t: head = (t/32)*2 + ((t%32)/16), edge = t%16
// Reductions over the 16 edges are __shfl_xor within the aligned 16-lane group.
// ---------------------------------------------------------------------------
__global__ __launch_bounds__(128)
void edge_attn(const float* __restrict__ qkvT,
               const int* __restrict__ col_ind,
               float* __restrict__ out) {
  const float* qT = qkvT;
  const float* kT = qkvT + (size_t)1 * NNODES * HID;
  const float* vT = qkvT + (size_t)2 * NNODES * HID;

  const int row  = blockIdx.x;
  const int t    = threadIdx.x;
  const int lane = t & 31;
  const int head = ((t >> 5) << 1) | (lane >> 4);  // 0..7
  const int e    = lane & 15;                      // 0..15

  const int col = col_ind[row * DEG + e];

  // score = sum_d q[row,d,head] * k[col,d,head]   (16 contiguous floats each)
  const float4* qf = (const float4*)(qT + ((size_t)head * NNODES + row) * HDIM);
  const float4* kf = (const float4*)(kT + ((size_t)head * NNODES + col) * HDIM);
  float s = 0.0f;
  #pragma unroll
  for (int i = 0; i < 4; ++i) {
    float4 a = qf[i], b = kf[i];
    s = fmaf(a.x, b.x, fmaf(a.y, b.y, fmaf(a.z, b.z, fmaf(a.w, b.w, s))));
  }

  // segment softmax over the 16 edges of this (row, head)
  float m = s;
  #pragma unroll
  for (int off = 1; off < 16; off <<= 1) m = fmaxf(m, __shfl_xor(m, off, 32));
  float ex = __expf(s - m);
  float den = ex;
  #pragma unroll
  for (int off = 1; off < 16; off <<= 1) den += __shfl_xor(den, off, 32);
  const float attn = ex / den;

  // out[row, d, head] = sum_e attn_e * v[col_e, d, head]
  const float4* vf = (const float4*)(vT + ((size_t)head * NNODES + col) * HDIM);
  #pragma unroll
  for (int i = 0; i < 4; ++i) {
    float4 vv = vf[i];
    float c0 = attn * vv.x, c1 = attn * vv.y, c2 = attn * vv.z, c3 = attn * vv.w;
    #pragma unroll
    for (int off = 1; off < 16; off <<= 1) {
      c0 += __shfl_xor(c0, off, 32);
      c1 += __shfl_xor(c1, off, 32);
      c2 += __shfl_xor(c2, off, 32);
      c3 += __shfl_xor(c3, off, 32);
    }
    if (e == 0) {
      const int d = i * 4;
      float* ob = out + (size_t)row * HID + head;
      ob[(d + 0) * NHEADS] = c0;
      ob[(d + 1) * NHEADS] = c1;
      ob[(d + 2) * NHEADS] = c2;
      ob[(d + 3) * NHEADS] = c3;
    }
  }
}

// ---------------------------------------------------------------------------
extern "C" void kernel_launch(void* const* d_in, const int* in_sizes, int n_in,
                              void* d_out, int out_size, void* d_ws, size_t ws_size,
                              hipStream_t stream) {
  const float* h    = (const float*)d_in[0];
  const float* Wq   = (const float*)d_in[1];
  const float* bq   = (const float*)d_in[2];
  const float* Wk   = (const float*)d_in[3];
  const float* bk   = (const float*)d_in[4];
  const float* Wv   = (const float*)d_in[5];
  const float* bv   = (const float*)d_in[6];
  // d_in[7] = row_ptr (unused: fixed degree 16), d_in[8] = col_ind, d_in[9] = num_heads
  const int* col_ind = (const int*)d_in[8];

  float* qkvT = (float*)d_ws;   // needs 3 * N * 128 * 4B = 76.8 MB

  dim3 gA(NTILES, 3, 1);
  qkv_project_wmma<<<gA, 32, 0, stream>>>(h, Wq, bq, Wk, bk, Wv, bv, qkvT);
  edge_attn<<<NNODES, 128, 0, stream>>>(qkvT, col_ind, (float*)d_out);
}